// ContrastiveLoss_76081050681843
// MI455X (gfx1250) — compile-verified
//
#include <hip/hip_runtime.h>

typedef __attribute__((ext_vector_type(2))) float v2f;
typedef __attribute__((ext_vector_type(8))) float v8f;

#define N 8192
#define D 256
#define KC 32        // K-chunk staged in LDS
#define LDSS 36      // padded LDS row stride in dwords (conflict-free)
#define TILE 128

// ---------------- per-row sums: s[i] = sum_k x[i,k], sq[i] = sum_k x[i,k]^2 ----------------
__global__ __launch_bounds__(256) void row_stats_kernel(const float* __restrict__ X,
                                                        float* __restrict__ rowsum,
                                                        float* __restrict__ rowsq) {
    const int row = blockIdx.x;
    const int tid = threadIdx.x;
    float v = X[(size_t)row * D + tid];
    float s = v, q = v * v;
#pragma unroll
    for (int off = 16; off >= 1; off >>= 1) {
        s += __shfl_xor(s, off, 32);
        q += __shfl_xor(q, off, 32);
    }
    __shared__ float ss[8], qq[8];
    const int wv = tid >> 5, ln = tid & 31;
    if (ln == 0) { ss[wv] = s; qq[wv] = q; }
    __syncthreads();
    if (tid == 0) {
        float ts = 0.f, tq = 0.f;
#pragma unroll
        for (int w = 0; w < 8; ++w) { ts += ss[w]; tq += qq[w]; }
        rowsum[row] = ts;
        rowsq[row]  = tq;
    }
}

__global__ void zero_kernel(float* acc) { acc[0] = 0.0f; }

// ---------------- fused Gram (f32 WMMA) + contrastive-loss epilogue + reduction ----------------
__global__ __launch_bounds__(256) void gram_loss_kernel(const float* __restrict__ X,
                                                        const long long* __restrict__ tgt,
                                                        const float* __restrict__ rowsum,
                                                        const float* __restrict__ rowsq,
                                                        float* __restrict__ acc_out) {
    __shared__ float As[TILE * LDSS];
    __shared__ float Bs[TILE * LDSS];

    const int tid    = threadIdx.x;
    const int wv     = tid >> 5;      // wave 0..7
    const int lane   = tid & 31;
    const int laneLo = lane & 15;
    const int laneHi = lane >> 4;     // 0 or 1
    const int iBase  = blockIdx.y * TILE;
    const int jBase  = blockIdx.x * TILE;

    v8f acc[8] = {};                  // wave's 16x128 strip: 8 tiles of 16x16 f32

    const float* gA = X + (size_t)iBase * D;
    const float* gB = X + (size_t)jBase * D;

    for (int kc = 0; kc < D; kc += KC) {
        // stage 128x32 A-chunk and B-chunk into LDS (float4 fills)
#pragma unroll
        for (int it = 0; it < 4; ++it) {
            int idx = it * 256 + tid;         // 0..1023
            int row = idx >> 3;               // 8 float4 per row
            int c4  = (idx & 7) << 2;         // dword offset within chunk
            float4 av = *reinterpret_cast<const float4*>(gA + (size_t)row * D + kc + c4);
            float4 bv = *reinterpret_cast<const float4*>(gB + (size_t)row * D + kc + c4);
            *reinterpret_cast<float4*>(As + row * LDSS + c4) = av;
            *reinterpret_cast<float4*>(Bs + row * LDSS + c4) = bv;
        }
        __syncthreads();

#pragma unroll
        for (int k0 = 0; k0 < KC; k0 += 4) {
            // A fragment: 16x4 f32; lane L<16 -> row wv*16+L, K=k0..k0+1; lane L+16 -> K=k0+2..k0+3
            v2f a = *reinterpret_cast<const v2f*>(As + (wv * 16 + laneLo) * LDSS + k0 + 2 * laneHi);
#pragma unroll
            for (int b = 0; b < 8; ++b) {
                // B = X^T for X@X^T: identical per-lane pattern as A, taken from the j-row block
                v2f bb = *reinterpret_cast<const v2f*>(Bs + (b * 16 + laneLo) * LDSS + k0 + 2 * laneHi);
                acc[b] = __builtin_amdgcn_wmma_f32_16x16x4_f32(
                    /*neg_a=*/false, a, /*neg_b=*/false, bb,
                    /*c_mod=*/(short)0, acc[b], /*reuse_a=*/false, /*reuse_b=*/false);
            }
        }
        __syncthreads();
    }

    // ---- fused epilogue ----
    // C/D layout: VGPR v holds M = v + 8*laneHi, N = laneLo
    const int iRow0 = iBase + wv * 16 + 8 * laneHi;
    long long ti[8]; float sqi[8], si[8];
#pragma unroll
    for (int v = 0; v < 8; ++v) {
        ti[v]  = tgt[iRow0 + v];
        sqi[v] = rowsq[iRow0 + v];
        si[v]  = rowsum[iRow0 + v];
    }

    const float EPS2TERM = (float)D * 1e-12f;   // d * eps^2
    float part = 0.0f;
#pragma unroll
    for (int b = 0; b < 8; ++b) {
        const int j = jBase + b * 16 + laneLo;
        const long long tj = tgt[j];
        const float sqj = rowsq[j], sj = rowsum[j];
#pragma unroll
        for (int v = 0; v < 8; ++v) {
            float g  = acc[b][v];
            float d2 = sqi[v] + sqj - 2.0f * g + 2e-6f * (si[v] - sj) + EPS2TERM;
            part += (ti[v] == tj) ? d2 : fmaxf(0.5f - d2, 0.0f);
        }
    }

    // wave reduce -> block reduce -> one atomic per block
#pragma unroll
    for (int off = 16; off >= 1; off >>= 1) part += __shfl_xor(part, off, 32);
    __shared__ float wsum[8];
    if (lane == 0) wsum[wv] = part;
    __syncthreads();
    if (tid == 0) {
        float t = 0.0f;
#pragma unroll
        for (int w = 0; w < 8; ++w) t += wsum[w];
        atomicAdd(acc_out, t);
    }
}

__global__ void finalize_kernel(const float* acc, float* out) {
    out[0] = acc[0] * (1.0f / (float)N);
}

extern "C" void kernel_launch(void* const* d_in, const int* in_sizes, int n_in,
                              void* d_out, int out_size, void* d_ws, size_t ws_size,
                              hipStream_t stream) {
    const float*      X   = (const float*)d_in[0];
    const long long*  tgt = (const long long*)d_in[1];
    float*            out = (float*)d_out;

    float* ws     = (float*)d_ws;
    float* accp   = ws;            // [1] scalar accumulator
    float* rowsum = ws + 64;       // [N]
    float* rowsq  = ws + 64 + N;   // [N]

    zero_kernel<<<1, 1, 0, stream>>>(accp);
    row_stats_kernel<<<N, 256, 0, stream>>>(X, rowsum, rowsq);
    gram_loss_kernel<<<dim3(N / TILE, N / TILE), 256, 0, stream>>>(X, tgt, rowsum, rowsq, accp);
    finalize_kernel<<<1, 1, 0, stream>>>(accp, out);
}